// MixGATLayer_14697378087233
// MI455X (gfx1250) — compile-verified
//
#include <hip/hip_runtime.h>
#include <hip/hip_bf16.h>
#include <math.h>

typedef float v2f __attribute__((ext_vector_type(2)));
typedef float v8f __attribute__((ext_vector_type(8)));

#define IN_DIM  128
#define HEADS   4
#define OUT_DIM 32
#define HD      128          // HEADS*OUT_DIM
#define LEAKY   0.2f
#define BETA_   0.5f
#define CMIX    1.2f

// ---------------------------------------------------------------------------
// K1: xp = x @ W  via V_WMMA_F32_16X16X4_F32 (fp32 end-to-end, matches ref).
// One wave per 16x16 output tile; block of 8 waves covers one 16-row strip
// across all 128 output columns. K loop: 32 iterations of K=4.
// A frag (16x4 f32): lane<16 -> row=lane, k-pair {k,k+1}; lane>=16 -> {k+2,k+3}
// B frag (4x16 f32): lane<16 -> col=lane, rows {k,k+1};   lane>=16 -> {k+2,k+3}
// D (16x16 f32):     vgpr r -> row = r + 8*(lane>=16), col = lane&15
// ---------------------------------------------------------------------------
__global__ void __launch_bounds__(256) gat_gemm_xp(const float* __restrict__ x,
                                                   const float* __restrict__ W,
                                                   float* __restrict__ xp)
{
    const int wave = threadIdx.x >> 5;     // 0..7 -> N tile
    const int lane = threadIdx.x & 31;
    const int half = lane >> 4;            // 0 or 1
    const int l15  = lane & 15;
    const int m0   = blockIdx.x * 16;
    const int n0   = wave * 16;

    const float* __restrict__ arow = x + (size_t)(m0 + l15) * IN_DIM + 2 * half;
    const float* __restrict__ bcol = W + (size_t)(2 * half) * HD + n0 + l15;

    v8f c = {};
    #pragma unroll 4
    for (int k = 0; k < IN_DIM; k += 4) {
        v2f a, b;
        a[0] = arow[k];
        a[1] = arow[k + 1];
        b[0] = bcol[(size_t)k * HD];
        b[1] = bcol[(size_t)k * HD + HD];
        c = __builtin_amdgcn_wmma_f32_16x16x4_f32(false, a, false, b,
                                                  (short)0, c, false, false);
    }

    float* __restrict__ drow = xp + (size_t)(m0 + 8 * half) * HD + n0 + l15;
    #pragma unroll
    for (int r = 0; r < 8; ++r)
        drow[(size_t)r * HD] = c[r];
}

// ---------------------------------------------------------------------------
// K2: per-(node,head) attention scores; init amax=alpha_self, denom=0;
// also zero the output accumulator (d_out).
// ---------------------------------------------------------------------------
__global__ void gat_scores_init(const float* __restrict__ xp,
                                const float* __restrict__ att_src,
                                const float* __restrict__ att_dst,
                                float* __restrict__ a_src,
                                float* __restrict__ a_dst,
                                float* __restrict__ alpha_self,
                                float* __restrict__ amax,
                                float* __restrict__ denom,
                                float* __restrict__ out,
                                int n_nodes)
{
    const int i = blockIdx.x * blockDim.x + threadIdx.x;
    if (i < n_nodes * HD) out[i] = 0.0f;
    if (i < n_nodes * HEADS) {
        const int n = i >> 2, h = i & 3;
        const float* __restrict__ row = xp + (size_t)n * HD + h * OUT_DIM;
        const float* __restrict__ as  = att_src + h * OUT_DIM;
        const float* __restrict__ ad  = att_dst + h * OUT_DIM;
        float sa = 0.0f, sd = 0.0f;
        #pragma unroll 8
        for (int d = 0; d < OUT_DIM; ++d) {
            const float v = row[d];
            sa += v * as[d];
            sd += v * ad[d];
        }
        a_src[i] = sa;
        a_dst[i] = sd;
        float al = sa + sd;
        al = (al > 0.0f) ? al : LEAKY * al;     // self-loop logit
        alpha_self[i] = al;
        amax[i]       = al;                      // self loop guarantees max >= this
        denom[i]      = 0.0f;
    }
}

// IEEE-order float atomic max via int punning (correct for +/- values)
__device__ __forceinline__ void atomicMaxF(float* addr, float val)
{
    if (val >= 0.0f) atomicMax((int*)addr, __float_as_int(val));
    else             atomicMin((unsigned int*)addr, __float_as_uint(val));
}

// ---------------------------------------------------------------------------
// K3: per-(edge,head) logit + segment-max into amax[dst].
// ---------------------------------------------------------------------------
__global__ void gat_edge_alpha(const long long* __restrict__ ei,
                               const float* __restrict__ a_src,
                               const float* __restrict__ a_dst,
                               float* __restrict__ ealpha,
                               float* __restrict__ amax,
                               int n_edges)
{
    const int t = blockIdx.x * blockDim.x + threadIdx.x;
    if (t >= n_edges * HEADS) return;
    const int e = t >> 2, h = t & 3;
    const int s = (int)ei[e];
    const int d = (int)ei[n_edges + e];
    float al = a_src[s * HEADS + h] + a_dst[d * HEADS + h];
    al = (al > 0.0f) ? al : LEAKY * al;
    ealpha[t] = al;
    atomicMaxF(&amax[d * HEADS + h], al);
}

// ---------------------------------------------------------------------------
// K4: per-(edge,head) exp(alpha - amax[dst]) + segment-sum into denom[dst].
// ---------------------------------------------------------------------------
__global__ void gat_edge_exp(const long long* __restrict__ ei,
                             const float* __restrict__ amax,
                             float* __restrict__ ealpha,
                             float* __restrict__ denom,
                             int n_edges)
{
    const int t = blockIdx.x * blockDim.x + threadIdx.x;
    if (t >= n_edges * HEADS) return;
    const int e = t >> 2, h = t & 3;
    const int d = (int)ei[n_edges + e];
    const float ex = expf(ealpha[t] - amax[d * HEADS + h]);
    ealpha[t] = ex;
    atomicAdd(&denom[d * HEADS + h], ex);
}

// ---------------------------------------------------------------------------
// K5: wave-per-edge weighted scatter-add:  out[dst] += e * xp[src]
// Lane L covers components [4L, 4L+3] (single head per lane), float4 gather
// of the 512B source row is fully coalesced and L2-resident on MI455X.
// ---------------------------------------------------------------------------
__global__ void __launch_bounds__(256) gat_edge_agg(const long long* __restrict__ ei,
                                                    const float* __restrict__ ealpha,
                                                    const float4* __restrict__ xp4,
                                                    float* __restrict__ out,
                                                    int n_edges)
{
    const int wave = (int)((blockIdx.x * (unsigned)blockDim.x + threadIdx.x) >> 5);
    const int lane = threadIdx.x & 31;
    if (wave >= n_edges) return;
    const int s = (int)ei[wave];
    const int d = (int)ei[n_edges + wave];
    const int h = lane >> 3;                          // 8 lanes per head
    const float  ex = ealpha[(size_t)wave * HEADS + h];
    const float4 v  = xp4[(size_t)s * 32 + lane];
    float* __restrict__ o = out + (size_t)d * HD + lane * 4;
    atomicAdd(o + 0, ex * v.x);
    atomicAdd(o + 1, ex * v.y);
    atomicAdd(o + 2, ex * v.z);
    atomicAdd(o + 3, ex * v.w);
}

// ---------------------------------------------------------------------------
// K6: fold in self-loop term, normalize, add bias, beta-mix swish.
// ---------------------------------------------------------------------------
__global__ void gat_finalize(const float* __restrict__ xp,
                             const float* __restrict__ alpha_self,
                             const float* __restrict__ amax,
                             const float* __restrict__ denom,
                             const float* __restrict__ bias,
                             float* __restrict__ out,
                             int n_nodes)
{
    const int i = blockIdx.x * blockDim.x + threadIdx.x;
    if (i >= n_nodes * HD) return;
    const int n  = i / HD;
    const int c  = i - n * HD;
    const int nh = n * HEADS + (c >> 5);
    const float es  = expf(alpha_self[nh] - amax[nh]);
    const float den = denom[nh] + es + 1e-16f;
    float z = (out[i] + es * xp[i]) / den + bias[c];
    const float sg = 1.0f / (1.0f + expf(-z));
    out[i] = BETA_ * z + (CMIX - BETA_) * (z * sg);
}

// ---------------------------------------------------------------------------
extern "C" void kernel_launch(void* const* d_in, const int* in_sizes, int n_in,
                              void* d_out, int out_size, void* d_ws, size_t ws_size,
                              hipStream_t stream)
{
    const float*     x        = (const float*)d_in[0];
    const long long* ei       = (const long long*)d_in[1];   // int64 [2, E]
    const float*     W        = (const float*)d_in[2];
    const float*     att_src  = (const float*)d_in[3];
    const float*     att_dst  = (const float*)d_in[4];
    const float*     bias     = (const float*)d_in[5];
    float*           out      = (float*)d_out;

    const int n_nodes = in_sizes[0] / IN_DIM;      // 50000
    const int n_edges = in_sizes[1] / 2;           // 1600000

    // workspace carve (floats); total ~55 MB
    float* ws         = (float*)d_ws;
    float* xp         = ws;                                   // n_nodes*HD
    float* a_src      = xp     + (size_t)n_nodes * HD;        // n_nodes*HEADS
    float* a_dst      = a_src  + (size_t)n_nodes * HEADS;
    float* alpha_self = a_dst  + (size_t)n_nodes * HEADS;
    float* amax       = alpha_self + (size_t)n_nodes * HEADS;
    float* denom      = amax   + (size_t)n_nodes * HEADS;
    float* ealpha     = denom  + (size_t)n_nodes * HEADS;     // n_edges*HEADS

    // K1: projection GEMM (WMMA). 50000 % 16 == 0 -> exact tiling, EXEC all-1s.
    gat_gemm_xp<<<n_nodes / 16, 256, 0, stream>>>(x, W, xp);

    // K2: scores + init (also zeroes d_out accumulator)
    {
        const int nthr = n_nodes * HD;
        gat_scores_init<<<(nthr + 255) / 256, 256, 0, stream>>>(
            xp, att_src, att_dst, a_src, a_dst, alpha_self, amax, denom, out, n_nodes);
    }

    // K3: logits + segment max
    {
        const int nthr = n_edges * HEADS;
        gat_edge_alpha<<<(nthr + 255) / 256, 256, 0, stream>>>(
            ei, a_src, a_dst, ealpha, amax, n_edges);
    }

    // K4: exp + segment sum
    {
        const int nthr = n_edges * HEADS;
        gat_edge_exp<<<(nthr + 255) / 256, 256, 0, stream>>>(
            ei, amax, ealpha, denom, n_edges);
    }

    // K5: weighted scatter-add aggregation (wave per edge)
    gat_edge_agg<<<(n_edges + 7) / 8, 256, 0, stream>>>(
        ei, ealpha, (const float4*)xp, out, n_edges);

    // K6: self-loop fold-in, normalize, bias, beta-mix swish
    {
        const int nthr = n_nodes * HD;
        gat_finalize<<<(nthr + 255) / 256, 256, 0, stream>>>(
            xp, alpha_self, amax, denom, bias, out, n_nodes);
    }
}